// Full_Post_Prob_85014582657568
// MI455X (gfx1250) — compile-verified
//
#include <hip/hip_runtime.h>
#include <hip/hip_bf16.h>
#include <math.h>

// ---------------- problem constants ----------------
#define BIMG   8
#define NPTS   512
#define KGRID  64
#define GSZ    4096          // KGRID*KGRID
#define MSEL   100
#define MPAD   112           // 7*16, WMMA-padded M
#define TWO_PI_F 6.283185307179586f
// s1 = ALPHA2+SIGMA2 = 128 ; s2 = ALPHA2+SIGMA2/2 = 96 ; 2*SIGMA2 = 128
#define E1C  (-0.5f / 128.0f)
#define E2C  (-0.5f / 96.0f)
#define G1C  (-0.5f / 128.0f)

typedef __attribute__((ext_vector_type(2))) float v2f;
typedef __attribute__((ext_vector_type(8))) float v8f;

// ---------------- output layout (floats) ----------------
#define OUT_M   ((size_t)0)                         // 8*512*4096 = 16777216
#define OUT_V   ((size_t)16777216)                  // 8*4096    = 32768
#define OUT_A   ((size_t)(16777216 + 32768))        // 8*100*100 = 80000
#define OUT_BM  (OUT_A + 80000)                     // 80000
#define OUT_MI  (OUT_BM + 80000)                    // 800

// =====================================================================
// K1: per-point separable factors + stacked GEMM operands
//   MX/MY  [b][p][k]          : exp(-dx^2/2s1)/sqrt(2pi s1)
//   SA     [b] 64x1024  AL[iy][p]=TY', AL[iy][512+p]=-MY^2
//   SB     [b] 1024x64  BR[p][ix]=TX', BR[512+p][ix]= MX^2
//   EXT    [b] 128x512  AL[a][p] = exp(-(x-t_a)^2/2s2)/sqrt(2pi s2) (a<127)
//   EY     [b] 512x128  BR[p][a] =  same with y
// =====================================================================
__global__ void factors_kernel(const float* __restrict__ pts,
                               float* __restrict__ MX, float* __restrict__ MY,
                               float* __restrict__ SA, float* __restrict__ SB,
                               float* __restrict__ EXT, float* __restrict__ EY) {
    int gid = blockIdx.x * blockDim.x + threadIdx.x;
    if (gid >= BIMG * NPTS) return;
    int b = gid / NPTS, p = gid % NPTS;
    float x = pts[(size_t)(b * NPTS + p) * 2 + 0];
    float y = pts[(size_t)(b * NPTS + p) * 2 + 1];

    const float RS1 = rsqrtf(TWO_PI_F * 128.0f);
    const float RS2 = rsqrtf(TWO_PI_F * 96.0f);
    const float CS2 = rsqrtf(TWO_PI_F * 96.0f * TWO_PI_F * 128.0f); // sqrt(1/((2pi s2)(4pi sig2)))

    float* mx = MX + (size_t)(b * NPTS + p) * KGRID;
    float* my = MY + (size_t)(b * NPTS + p) * KGRID;
    float* sa = SA + (size_t)b * 64 * 1024;
    float* sb = SB + (size_t)b * 1024 * 64;

    for (int k = 0; k < KGRID; ++k) {
        float c  = 8.0f * k + 4.0f;
        float dx = x - c, dy = y - c;
        float mxv = __expf(E1C * dx * dx) * RS1;
        float myv = __expf(E1C * dy * dy) * RS1;
        float txv = __expf(E2C * dx * dx) * CS2;
        float tyv = __expf(E2C * dy * dy) * CS2;
        mx[k] = mxv;  my[k] = myv;
        sa[k * 1024 + p]        = tyv;
        sa[k * 1024 + 512 + p]  = -myv * myv;
        sb[p * 64 + k]          = txv;
        sb[(512 + p) * 64 + k]  = mxv * mxv;
    }
    float* ext = EXT + (size_t)b * 128 * 512;
    float* ey  = EY  + (size_t)b * 512 * 128;
    for (int a = 0; a < 128; ++a) {
        float t = 4.0f * a + 4.0f;
        float ddx = x - t, ddy = y - t;
        float exv = (a < 127) ? __expf(E2C * ddx * ddx) * RS2 : 0.0f;
        float eyv = (a < 127) ? __expf(E2C * ddy * ddy) * RS2 : 0.0f;
        ext[a * 512 + p]  = exv;
        ey [p * 128 + a]  = eyv;
    }
}

// =====================================================================
// K2: m = outer(MY, MX)  -> 67MB coalesced float4 stores (bandwidth bound)
// =====================================================================
__global__ void m_kernel(const float* __restrict__ MX, const float* __restrict__ MY,
                         float* __restrict__ outm) {
    int gid = blockIdx.x * 256 + threadIdx.x;       // B*N*64*16 = 4194304
    int ix4 = gid & 15;  int t = gid >> 4;
    int iy  = t & 63;    t >>= 6;
    int p   = t & 511;   int b = t >> 9;
    float myv = MY[(size_t)(b * NPTS + p) * KGRID + iy];
    float4 mx4 = *(const float4*)&MX[(size_t)(b * NPTS + p) * KGRID + ix4 * 4];
    float4 r;
    r.x = myv * mx4.x; r.y = myv * mx4.y; r.z = myv * mx4.z; r.w = myv * mx4.w;
    *(float4*)&outm[(size_t)(b * NPTS + p) * GSZ + iy * 64 + ix4 * 4] = r;
}

// =====================================================================
// K3: v = SA^T-stack GEMM  (64 x 1024 x 64 per image) via V_WMMA_F32_16X16X4_F32
// =====================================================================
__global__ void vgemm_kernel(const float* __restrict__ SA, const float* __restrict__ SB,
                             float* __restrict__ vraw, float* __restrict__ outv) {
    int b    = blockIdx.x;
    int wave = threadIdx.x >> 5;
    int lane = threadIdx.x & 31;
    int hf   = lane >> 4;
    int l15  = lane & 15;
    const float* AL = SA + (size_t)b * 64 * 1024;   // [i][k]
    const float* BR = SB + (size_t)b * 1024 * 64;   // [k][j]
    for (int tile = wave; tile < 16; tile += 8) {
        int i0 = (tile >> 2) * 16, j0 = (tile & 3) * 16;
        int m = i0 + l15, n = j0 + l15;
        v8f acc = {};
        for (int k = 0; k < 1024; k += 4) {
            int kk = k + 2 * hf;
            v2f a, bb;
            a.x  = AL[m * 1024 + kk];
            a.y  = AL[m * 1024 + kk + 1];
            bb.x = BR[kk * 64 + n];
            bb.y = BR[(kk + 1) * 64 + n];
            acc = __builtin_amdgcn_wmma_f32_16x16x4_f32(false, a, false, bb,
                                                        (short)0, acc, false, false);
        }
        for (int r = 0; r < 8; ++r) {
            int row = i0 + r + 8 * hf;
            int g   = row * 64 + n;
            float val = acc[r];
            vraw[b * GSZ + g] = val;
            outv[b * GSZ + g] = fmaxf(val, 1e-6f);
        }
    }
}

// =====================================================================
// K4: top-M selection per image (bitonic in LDS), index sort, vM
// =====================================================================
__global__ void topm_kernel(const float* __restrict__ vraw,
                            int* __restrict__ minds, float* __restrict__ vm,
                            float* __restrict__ out_minds) {
    __shared__ float sv[4096];
    __shared__ int   si[4096];
    __shared__ int   stmp[128];
    int b = blockIdx.x, tid = threadIdx.x;          // 256 threads
    for (int i = tid; i < 4096; i += 256) { sv[i] = vraw[b * GSZ + i]; si[i] = i; }
    __syncthreads();
    // bitonic sort, "before" = descending value, ascending index on ties
    for (int ksz = 2; ksz <= 4096; ksz <<= 1) {
        for (int j = ksz >> 1; j > 0; j >>= 1) {
            for (int t = tid; t < 2048; t += 256) {
                int i = 2 * t - (t & (j - 1));
                int r = i + j;
                bool dirAsc = ((i & ksz) == 0);
                float vl = sv[i], vr = sv[r];
                int   il = si[i], ir = si[r];
                bool rBefore = (vr > vl) || (vr == vl && ir < il);
                if (dirAsc == rBefore) { sv[i] = vr; sv[r] = vl; si[i] = ir; si[r] = il; }
            }
            __syncthreads();
        }
    }
    for (int i = tid; i < 128; i += 256) stmp[i] = (i < MSEL) ? si[i] : 0x7FFFFFFF;
    __syncthreads();
    for (int ksz = 2; ksz <= 128; ksz <<= 1) {
        for (int j = ksz >> 1; j > 0; j >>= 1) {
            if (tid < 64) {
                int i = 2 * tid - (tid & (j - 1));
                int r = i + j;
                bool dirAsc = ((i & ksz) == 0);
                int a = stmp[i], c = stmp[r];
                if ((a > c) == dirAsc) { stmp[i] = c; stmp[r] = a; }
            }
            __syncthreads();
        }
    }
    if (tid < MSEL) {
        int g = stmp[tid];
        minds[b * 128 + tid]      = g;
        out_minds[b * MSEL + tid] = (float)g;
        vm[b * MPAD + tid]        = fmaxf(vraw[b * GSZ + g], 1e-6f) + 1e-10f;
    }
}

// =====================================================================
// K5: gather tmpm2 (512 x 112, zero-padded) from separable factors
// =====================================================================
__global__ void tm_kernel(const float* __restrict__ MX, const float* __restrict__ MY,
                          const int* __restrict__ minds, float* __restrict__ TM) {
    int gid = blockIdx.x * blockDim.x + threadIdx.x; // B*512*112 = 458752
    if (gid >= BIMG * NPTS * MPAD) return;
    int i = gid % MPAD; int t = gid / MPAD;
    int p = t % NPTS;   int b = t / NPTS;
    float val = 0.0f;
    if (i < MSEL) {
        int g  = minds[b * 128 + i];
        int iy = g >> 6, ix = g & 63;
        val = MY[(size_t)(b * NPTS + p) * KGRID + iy] *
              MX[(size_t)(b * NPTS + p) * KGRID + ix];
    }
    TM[(size_t)(b * NPTS + p) * MPAD + i] = val;
}

// =====================================================================
// K6: midpoint Gaussian GEMM  H = EXT * EY  (128 x 512 x 128 per image)
// =====================================================================
__global__ void hgemm_kernel(const float* __restrict__ EXT, const float* __restrict__ EY,
                             float* __restrict__ H) {
    int b    = blockIdx.x;
    int wave = threadIdx.x >> 5;
    int lane = threadIdx.x & 31;
    int hf   = lane >> 4, l15 = lane & 15;
    const float* AL = EXT + (size_t)b * 128 * 512;  // [a][p]
    const float* BR = EY  + (size_t)b * 512 * 128;  // [p][a]
    for (int tile = wave; tile < 64; tile += 8) {
        int i0 = (tile >> 3) * 16, j0 = (tile & 7) * 16;
        int m = i0 + l15, n = j0 + l15;
        v8f acc = {};
        for (int k = 0; k < 512; k += 4) {
            int kk = k + 2 * hf;
            v2f a, bb;
            a.x  = AL[m * 512 + kk];
            a.y  = AL[m * 512 + kk + 1];
            bb.x = BR[kk * 128 + n];
            bb.y = BR[(kk + 1) * 128 + n];
            acc = __builtin_amdgcn_wmma_f32_16x16x4_f32(false, a, false, bb,
                                                        (short)0, acc, false, false);
        }
        for (int r = 0; r < 8; ++r) {
            int row = i0 + r + 8 * hf;
            H[(size_t)b * 128 * 128 + row * 128 + n] = acc[r];
        }
    }
}

// =====================================================================
// K7: P = tmpm2^T * tmpm2  (112 x 512 x 112 per image)
// =====================================================================
__global__ void pgemm_kernel(const float* __restrict__ TM, float* __restrict__ P) {
    int b    = blockIdx.x;
    int wave = threadIdx.x >> 5;
    int lane = threadIdx.x & 31;
    int hf   = lane >> 4, l15 = lane & 15;
    const float* T = TM + (size_t)b * NPTS * MPAD;  // [p][i]
    for (int tile = wave; tile < 49; tile += 8) {
        int i0 = (tile / 7) * 16, j0 = (tile % 7) * 16;
        int m = i0 + l15, n = j0 + l15;
        v8f acc = {};
        for (int k = 0; k < 512; k += 4) {
            int kk = k + 2 * hf;
            v2f a, bb;
            a.x  = T[kk * MPAD + m];
            a.y  = T[(kk + 1) * MPAD + m];
            bb.x = T[kk * MPAD + n];
            bb.y = T[(kk + 1) * MPAD + n];
            acc = __builtin_amdgcn_wmma_f32_16x16x4_f32(false, a, false, bb,
                                                        (short)0, acc, false, false);
        }
        for (int r = 0; r < 8; ++r) {
            int row = i0 + r + 8 * hf;
            P[(size_t)b * MPAD * MPAD + row * MPAD + n] = acc[r];
        }
    }
}

// =====================================================================
// K8: assemble A[i][j] = g1*(H[sx][sy] + N) - P[i][j] ; diag = 1e-10
// =====================================================================
__global__ void a_kernel(const float* __restrict__ H, const float* __restrict__ P,
                         const int* __restrict__ minds, float* __restrict__ outA) {
    int gid = blockIdx.x * blockDim.x + threadIdx.x;
    if (gid >= BIMG * MSEL * MSEL) return;
    int j = gid % MSEL; int t = gid / MSEL;
    int i = t % MSEL;   int b = t / MSEL;
    if (i == j) { outA[(size_t)b * MSEL * MSEL + i * MSEL + j] = 1e-10f; return; }
    int gi = minds[b * 128 + i], gj = minds[b * 128 + j];
    int ixi = gi & 63, iyi = gi >> 6;
    int ixj = gj & 63, iyj = gj >> 6;
    float ddx = 8.0f * (float)(ixi - ixj);
    float ddy = 8.0f * (float)(iyi - iyj);
    float g1 = __expf(G1C * (ddx * ddx + ddy * ddy)) * (1.0f / (TWO_PI_F * 128.0f));
    int sx = ixi + ixj, sy = iyi + iyj;
    float hv = H[(size_t)b * 128 * 128 + sx * 128 + sy];
    float av = g1 * hv + 512.0f * g1 - P[(size_t)b * MPAD * MPAD + i * MPAD + j];
    outA[(size_t)b * MSEL * MSEL + i * MSEL + j] = av;
}

// =====================================================================
// K9: two Gauss-Jordan inversions (partial pivoting) per image in LDS
// =====================================================================
#define AUGSTR 201
__device__ void gj_invert(float* aug, float* fac, float* redv, int* redi, int tid) {
    for (int col = 0; col < MSEL; ++col) {
        // partial pivot search
        float bv = -1.0f; int bi = col;
        for (int r = col + tid; r < MSEL; r += 256) {
            float av = fabsf(aug[r * AUGSTR + col]);
            if (av > bv) { bv = av; bi = r; }
        }
        redv[tid] = bv; redi[tid] = bi;
        __syncthreads();
        for (int s = 128; s > 0; s >>= 1) {
            if (tid < s && redv[tid + s] > redv[tid]) {
                redv[tid] = redv[tid + s]; redi[tid] = redi[tid + s];
            }
            __syncthreads();
        }
        int piv = redi[0];
        float pinv = 1.0f / aug[piv * AUGSTR + col];
        __syncthreads();
        // swap rows (col <-> piv) and scale pivot row
        for (int c = tid; c < 200; c += 256) {
            float a = aug[col * AUGSTR + c];
            float p = aug[piv * AUGSTR + c];
            aug[col * AUGSTR + c] = p * pinv;
            if (piv != col) aug[piv * AUGSTR + c] = a;
        }
        __syncthreads();
        for (int r = tid; r < MSEL; r += 256) fac[r] = aug[r * AUGSTR + col];
        __syncthreads();
        for (int t2 = tid; t2 < MSEL * 200; t2 += 256) {
            int r = t2 / 200, c = t2 % 200;
            if (r != col) aug[r * AUGSTR + c] -= fac[r] * aug[col * AUGSTR + c];
        }
        __syncthreads();
    }
}

__global__ void inv_kernel(const float* __restrict__ Ain, const float* __restrict__ vm,
                           float* __restrict__ Bm) {
    extern __shared__ float smem[];
    float* aug  = smem;                       // 100 x 201
    float* fac  = smem + MSEL * AUGSTR;       // 104
    float* redv = fac + 104;                  // 256
    int*   redi = (int*)(redv + 256);         // 256
    int b = blockIdx.x, tid = threadIdx.x;
    // [A | I]
    for (int t = tid; t < MSEL * 200; t += 256) {
        int r = t / 200, c = t % 200;
        float v = (c < MSEL) ? Ain[(size_t)b * MSEL * MSEL + r * MSEL + c]
                             : ((c - MSEL) == r ? 1.0f : 0.0f);
        aug[r * AUGSTR + c] = v;
    }
    __syncthreads();
    gj_invert(aug, fac, redv, redi, tid);     // right half = inv(A)
    // tmpA = vM_i * invA_ij * vM_j + diag(vM) -> left half (reads right half only)
    for (int t = tid; t < MSEL * MSEL; t += 256) {
        int r = t / MSEL, c = t % MSEL;
        float v = vm[b * MPAD + r] * aug[r * AUGSTR + MSEL + c] * vm[b * MPAD + c];
        if (r == c) v += vm[b * MPAD + r];
        aug[r * AUGSTR + c] = v;
    }
    __syncthreads();
    for (int t = tid; t < MSEL * MSEL; t += 256) {
        int r = t / MSEL, c = t % MSEL;
        aug[r * AUGSTR + MSEL + c] = (r == c) ? 1.0f : 0.0f;
    }
    __syncthreads();
    gj_invert(aug, fac, redv, redi, tid);     // right half = Bm
    for (int t = tid; t < MSEL * MSEL; t += 256) {
        int r = t / MSEL, c = t % MSEL;
        Bm[(size_t)b * MSEL * MSEL + r * MSEL + c] = aug[r * AUGSTR + MSEL + c];
    }
}

// =====================================================================
// launch
// =====================================================================
extern "C" void kernel_launch(void* const* d_in, const int* in_sizes, int n_in,
                              void* d_out, int out_size, void* d_ws, size_t ws_size,
                              hipStream_t stream) {
    (void)in_sizes; (void)n_in; (void)out_size; (void)ws_size;
    const float* pts = (const float*)d_in[0];   // (8,512,2) f32 ; st_sizes unused

    float* ws = (float*)d_ws;
    size_t o = 0;
    float* MX  = ws + o; o += (size_t)BIMG * NPTS * KGRID;   // 262144
    float* MY  = ws + o; o += (size_t)BIMG * NPTS * KGRID;
    float* SA  = ws + o; o += (size_t)BIMG * 64 * 1024;      // 524288
    float* SB  = ws + o; o += (size_t)BIMG * 1024 * 64;
    float* EXT = ws + o; o += (size_t)BIMG * 128 * 512;
    float* EY  = ws + o; o += (size_t)BIMG * 512 * 128;
    float* H   = ws + o; o += (size_t)BIMG * 128 * 128;
    float* TM  = ws + o; o += (size_t)BIMG * NPTS * MPAD;
    float* P   = ws + o; o += (size_t)BIMG * MPAD * MPAD;
    float* VR  = ws + o; o += (size_t)BIMG * GSZ;
    float* VM  = ws + o; o += (size_t)BIMG * MPAD;
    int* MINDS = (int*)(ws + o); o += (size_t)BIMG * 128;

    float* out   = (float*)d_out;
    float* out_m = out + OUT_M;
    float* out_v = out + OUT_V;
    float* out_A = out + OUT_A;
    float* out_B = out + OUT_BM;
    float* out_I = out + OUT_MI;

    factors_kernel<<<(BIMG * NPTS + 255) / 256, 256, 0, stream>>>(pts, MX, MY, SA, SB, EXT, EY);
    m_kernel<<<(BIMG * NPTS * KGRID * 16) / 256, 256, 0, stream>>>(MX, MY, out_m);
    vgemm_kernel<<<BIMG, 256, 0, stream>>>(SA, SB, VR, out_v);
    topm_kernel<<<BIMG, 256, 0, stream>>>(VR, MINDS, VM, out_I);
    tm_kernel<<<(BIMG * NPTS * MPAD + 255) / 256, 256, 0, stream>>>(MX, MY, MINDS, TM);
    hgemm_kernel<<<BIMG, 256, 0, stream>>>(EXT, EY, H);
    pgemm_kernel<<<BIMG, 256, 0, stream>>>(TM, P);
    a_kernel<<<(BIMG * MSEL * MSEL + 255) / 256, 256, 0, stream>>>(H, P, MINDS, out_A);

    size_t smem_bytes = (size_t)(MSEL * AUGSTR + 104 + 256 + 256) * sizeof(float);
    inv_kernel<<<BIMG, 256, smem_bytes, stream>>>(out_A, VM, out_B);
}